// GCNNet_Cora_34832184770970
// MI455X (gfx1250) — compile-verified
//
#include <hip/hip_runtime.h>

typedef __attribute__((ext_vector_type(2))) float v2f;
typedef __attribute__((ext_vector_type(8))) float v8f;

// ---------------- degree / normalization ----------------
__global__ void k_deg_init(int* __restrict__ deg, int n) {
    int i = blockIdx.x * blockDim.x + threadIdx.x;
    if (i < n) deg[i] = 1;                       // self-loop contributes 1
}

__global__ void k_deg_edges(const int* __restrict__ dst, int e, int* __restrict__ deg) {
    int i = blockIdx.x * blockDim.x + threadIdx.x;
    if (i < e) atomicAdd(&deg[dst[i]], 1);
}

__global__ void k_dinv(const int* __restrict__ deg, float* __restrict__ dinv, int n) {
    int i = blockIdx.x * blockDim.x + threadIdx.x;
    if (i < n) dinv[i] = rsqrtf((float)deg[i]);  // deg >= 1 always
}

// ---------------- GEMM1: h0 = x @ W1  (N x 512 x 128), fp32 WMMA ----------------
// 4 waves/block; wave w computes a 16x32 slab (two 16x16 tiles, independent
// accumulator chains for ILP). Block covers all 128 output columns.
__global__ void k_gemm1_wmma(const float* __restrict__ x,
                             const float* __restrict__ W1,
                             float* __restrict__ h0, int n) {
    const int lane = threadIdx.x & 31;
    const int wave = threadIdx.x >> 5;
    const int row0 = blockIdx.x * 16;
    const int col0 = wave * 32;                  // 4 waves * 32 = 128 cols
    const int m  = lane & 15;
    const int kh = (lane >> 4) << 1;             // 0 for lanes 0-15, 2 for lanes 16-31
    const int rowA = min(row0 + m, n - 1);
    const float* __restrict__ xrow = x + (size_t)rowA * 512;

    v8f acc0 = {}, acc1 = {};
#pragma unroll 4
    for (int k = 0; k < 512; k += 4) {
        v2f a;
        a.x = xrow[k + kh];
        a.y = xrow[k + kh + 1];
        const float* __restrict__ w0 = W1 + (size_t)(k + kh)     * 128 + col0 + m;
        const float* __restrict__ w1 = W1 + (size_t)(k + kh + 1) * 128 + col0 + m;
        v2f b0, b1;
        b0.x = w0[0];  b0.y = w1[0];
        b1.x = w0[16]; b1.y = w1[16];
        acc0 = __builtin_amdgcn_wmma_f32_16x16x4_f32(false, a, false, b0,
                                                     (short)0, acc0, false, false);
        acc1 = __builtin_amdgcn_wmma_f32_16x16x4_f32(false, a, false, b1,
                                                     (short)0, acc1, false, false);
    }
    const int rowD0 = row0 + ((lane >> 4) << 3); // +8 for upper lane half
    float* __restrict__ o0 = h0 + (size_t)rowD0 * 128 + col0 + m;
    if (row0 + 16 <= n) {                        // wave-uniform fast path
#pragma unroll
        for (int r = 0; r < 8; ++r) {
            o0[(size_t)r * 128]      = acc0[r];
            o0[(size_t)r * 128 + 16] = acc1[r];
        }
    } else {
#pragma unroll
        for (int r = 0; r < 8; ++r) {
            if (rowD0 + r < n) {
                o0[(size_t)r * 128]      = acc0[r];
                o0[(size_t)r * 128 + 16] = acc1[r];
            }
        }
    }
}

// ---------------- self-loop init (float4): out = dinv^2 * h ----------------
// one thread per float4; lgQ = log2(F/4) quads per node
__global__ void k_selfloop4(const float4* __restrict__ h, const float* __restrict__ dinv,
                            float4* __restrict__ out, int nq, int lgQ) {
    int i = blockIdx.x * blockDim.x + threadIdx.x;
    if (i < nq) {
        float d = dinv[i >> lgQ];
        float s = d * d;
        float4 v = h[i];
        v.x *= s; v.y *= s; v.z *= s; v.w *= s;
        out[i] = v;
    }
}

// ---------------- edge aggregation, 128 features: one wave per edge ----------------
__global__ void k_agg128(const int* __restrict__ src, const int* __restrict__ dst,
                         const float* __restrict__ dinv,
                         const float* __restrict__ h, float* __restrict__ out, int e) {
    int edge = blockIdx.x * (blockDim.x >> 5) + (threadIdx.x >> 5);
    if (edge >= e) return;
    int lane = threadIdx.x & 31;
    int s = src[edge], d = dst[edge];
    float norm = dinv[s] * dinv[d];
    float4 v = ((const float4*)(h + (size_t)s * 128))[lane];   // global_load_b128
    float* op = out + (size_t)d * 128 + lane * 4;
    atomicAdd(op + 0, norm * v.x);
    atomicAdd(op + 1, norm * v.y);
    atomicAdd(op + 2, norm * v.z);
    atomicAdd(op + 3, norm * v.w);
}

// ---------------- GEMM2: h2 = relu(t1 + b1) @ W2  (N x 128 x 16), fp32 WMMA ----------
// 2 waves/block; K split into two independent accumulator chains.
__global__ void k_gemm2_wmma(const float* __restrict__ t1,
                             const float* __restrict__ b1,
                             const float* __restrict__ W2,
                             float* __restrict__ h2, int n) {
    const int lane = threadIdx.x & 31;
    const int wave = threadIdx.x >> 5;
    const int row0 = (blockIdx.x * 2 + wave) * 16;
    const int m  = lane & 15;
    const int kh = (lane >> 4) << 1;
    const int rowA = min(row0 + m, n - 1);
    const float* __restrict__ trow = t1 + (size_t)rowA * 128;

    v8f acc0 = {}, acc1 = {};
#pragma unroll 4
    for (int k = 0; k < 64; k += 4) {
        int ka = k + kh, kb = k + 64 + kh;
        v2f a0, a1, b0, b1;
        a0.x = fmaxf(trow[ka]     + b1[ka],     0.0f);   // fused bias+relu
        a0.y = fmaxf(trow[ka + 1] + b1[ka + 1], 0.0f);
        a1.x = fmaxf(trow[kb]     + b1[kb],     0.0f);
        a1.y = fmaxf(trow[kb + 1] + b1[kb + 1], 0.0f);
        b0.x = W2[ka * 16 + m];
        b0.y = W2[(ka + 1) * 16 + m];
        b1.x = W2[kb * 16 + m];
        b1.y = W2[(kb + 1) * 16 + m];
        acc0 = __builtin_amdgcn_wmma_f32_16x16x4_f32(false, a0, false, b0,
                                                     (short)0, acc0, false, false);
        acc1 = __builtin_amdgcn_wmma_f32_16x16x4_f32(false, a1, false, b1,
                                                     (short)0, acc1, false, false);
    }
    acc0 += acc1;                                // merge the two K-halves
    const int rowD0 = row0 + ((lane >> 4) << 3);
    float* __restrict__ o = h2 + (size_t)rowD0 * 16 + m;
    if (row0 + 16 <= n) {
#pragma unroll
        for (int r = 0; r < 8; ++r) o[r * 16] = acc0[r];
    } else {
#pragma unroll
        for (int r = 0; r < 8; ++r)
            if (rowD0 + r < n) o[r * 16] = acc0[r];
    }
}

// ---------------- edge aggregation, 16 features: 16 lanes per edge ----------------
__global__ void k_agg16(const int* __restrict__ src, const int* __restrict__ dst,
                        const float* __restrict__ dinv,
                        const float* __restrict__ h, float* __restrict__ out, int e) {
    int t = blockIdx.x * blockDim.x + threadIdx.x;   // e*16 = 25.6M fits in int
    if (t >= e * 16) return;
    int edge = t >> 4;
    int f    = t & 15;
    int s = src[edge], d = dst[edge];
    float norm = dinv[s] * dinv[d];
    atomicAdd(&out[(size_t)d * 16 + f], norm * h[(size_t)s * 16 + f]);
}

// ---------------- head: logits = relu(t2+b2) @ Wfc + bfc; log_softmax ----------------
__global__ void k_head(const float* __restrict__ t2, const float* __restrict__ b2,
                       const float* __restrict__ Wfc, const float* __restrict__ bfc,
                       float* __restrict__ out, int n) {
    __shared__ float sW[16 * 40];
    __shared__ float sb[40];
    for (int i = threadIdx.x; i < 640; i += blockDim.x) sW[i] = Wfc[i];
    for (int i = threadIdx.x; i < 40;  i += blockDim.x) sb[i] = bfc[i];
    __syncthreads();
    int node = blockIdx.x * blockDim.x + threadIdx.x;
    if (node >= n) return;

    float h[16];
#pragma unroll
    for (int f = 0; f < 16; ++f)
        h[f] = fmaxf(t2[(size_t)node * 16 + f] + b2[f], 0.0f);

    float logit[40];
    float mx = -3.4e38f;
#pragma unroll
    for (int c = 0; c < 40; ++c) {
        float acc = sb[c];
#pragma unroll
        for (int f = 0; f < 16; ++f) acc += h[f] * sW[f * 40 + c];
        logit[c] = acc;
        mx = fmaxf(mx, acc);
    }
    float sum = 0.0f;
#pragma unroll
    for (int c = 0; c < 40; ++c) sum += expf(logit[c] - mx);
    float lse = mx + logf(sum);
#pragma unroll
    for (int c = 0; c < 40; ++c) out[(size_t)node * 40 + c] = logit[c] - lse;
}

// ---------------- host-side orchestration ----------------
extern "C" void kernel_launch(void* const* d_in, const int* in_sizes, int n_in,
                              void* d_out, int out_size, void* d_ws, size_t ws_size,
                              hipStream_t stream) {
    const float* x   = (const float*)d_in[0];
    const int*   ei  = (const int*)  d_in[1];
    const float* W1  = (const float*)d_in[2];
    const float* b1  = (const float*)d_in[3];
    const float* W2  = (const float*)d_in[4];
    const float* b2  = (const float*)d_in[5];
    const float* Wfc = (const float*)d_in[6];
    const float* bfc = (const float*)d_in[7];

    const int n = in_sizes[0] / 512;   // 100000
    const int e = in_sizes[1] / 2;     // 1600000
    const int* src = ei;
    const int* dst = ei + e;

    char* ws = (char*)d_ws;
    int*   deg  = (int*)ws;                 ws += (size_t)n * sizeof(int);
    float* dinv = (float*)ws;               ws += (size_t)n * sizeof(float);
    float* bufA = (float*)ws;               ws += (size_t)n * 128 * sizeof(float);
    float* bufB = (float*)ws;               // n*128 floats

    float* h0 = bufA;                       // stage 1 GEMM out
    float* t1 = bufB;                       // stage 1 agg out
    float* h2 = bufA;                       // stage 2 GEMM out (h0 dead)
    float* t2 = bufA + (size_t)n * 16;      // stage 2 agg out (disjoint from h2)
    float* out = (float*)d_out;

    // degrees + normalization
    k_deg_init <<<(n + 255) / 256, 256, 0, stream>>>(deg, n);
    k_deg_edges<<<(e + 255) / 256, 256, 0, stream>>>(dst, e, deg);
    k_dinv     <<<(n + 255) / 256, 256, 0, stream>>>(deg, dinv, n);

    // conv1
    k_gemm1_wmma<<<(n + 15) / 16, 128, 0, stream>>>(x, W1, h0, n);
    int nq1 = n * 32;                       // float4s for F=128
    k_selfloop4 <<<(nq1 + 255) / 256, 256, 0, stream>>>((const float4*)h0, dinv,
                                                        (float4*)t1, nq1, 5);
    k_agg128    <<<(e + 7) / 8, 256, 0, stream>>>(src, dst, dinv, h0, t1, e);

    // conv2 (relu+b1 fused into GEMM2 A-loads)
    k_gemm2_wmma<<<(n + 31) / 32, 64, 0, stream>>>(t1, b1, W2, h2, n);
    int nq2 = n * 4;                        // float4s for F=16
    k_selfloop4 <<<(nq2 + 255) / 256, 256, 0, stream>>>((const float4*)h2, dinv,
                                                        (float4*)t2, nq2, 2);
    k_agg16     <<<((e * 16) + 255) / 256, 256, 0, stream>>>(src, dst, dinv, h2, t2, e);

    // fc head + log_softmax (relu+b2 fused)
    k_head<<<(n + 255) / 256, 256, 0, stream>>>(t2, b2, Wfc, bfc, out, n);
}